// LBMGramSchmidtLayer_23639499997808
// MI455X (gfx1250) — compile-verified
//
#include <hip/hip_runtime.h>

typedef float v2f __attribute__((ext_vector_type(2)));
typedef float v8f __attribute__((ext_vector_type(8)));

#define NXL 1024
#define NYL 1024
#define TAU 0.6f
#define GYF (-1e-4f)

// A-matrix: A[k=moment p][q] = Minv[q][p] / tau = M[p][q] / norm2[p] / tau,
// padded to K=12 rows x 16 cols (zeros outside 9x9). Built at compile time.
struct ATab { float v[12][16]; };

static constexpr ATab makeAT() {
    const signed char M[9][9] = {
        { 1,  1,  1,  1,  1,  1,  1,  1,  1},   // rho
        { 0,  1,  0, -1,  0,  1, -1, -1,  1},   // jx
        { 0,  0,  1,  0, -1,  1,  1, -1, -1},   // jy
        {-4, -1, -1, -1, -1,  2,  2,  2,  2},   // e
        { 4, -2, -2, -2, -2,  1,  1,  1,  1},   // eps
        { 0, -2,  0,  2,  0,  1, -1, -1,  1},   // qx
        { 0,  0, -2,  0,  2,  1,  1, -1, -1},   // qy
        { 0,  1, -1,  1, -1,  0,  0,  0,  0},   // pxx
        { 0,  0,  0,  0,  0,  1, -1,  1, -1},   // pxy
    };
    const float nrm2[9] = {9.f, 6.f, 6.f, 36.f, 36.f, 12.f, 12.f, 4.f, 4.f};
    ATab t{};
    for (int k = 0; k < 12; ++k)
        for (int q = 0; q < 16; ++q) {
            float x = 0.0f;
            if (k < 9 && q < 9)
                x = (float)M[k][q] / nrm2[k] * (1.0f / TAU);
            t.v[k][q] = x;
        }
    return t;
}

__constant__ ATab g_AT = makeAT();

// One fused stream+collide step. 16 cells per wave32 (cells duplicated in the
// two half-waves to feed the WMMA B-matrix rows without cross-lane shuffles).
__global__ __launch_bounds__(256) void lbm_step(const float* __restrict__ fin,
                                                const float* __restrict__ w,
                                                float* __restrict__ fout) {
    const int tid  = blockIdx.x * blockDim.x + threadIdx.x;
    const int lane = threadIdx.x & 31;
    const int half = lane >> 4;          // 0 = lanes 0-15, 1 = lanes 16-31
    const int q16  = lane & 15;
    const int cell = (tid >> 5) * 16 + q16;   // 16 cells per wave
    const int x = cell >> 10;
    const int y = cell & (NYL - 1);

    // ---- periodic streaming: f[i] = fin[(x-ex)%NX][(y-ey)%NY][i] ----
    const int exv[9] = {0, 1, 0, -1, 0, 1, -1, -1, 1};
    const int eyv[9] = {0, 0, 1, 0, -1, 1, 1, -1, -1};
    float f[9];
#pragma unroll
    for (int i = 0; i < 9; ++i) {
        const int nx = (x - exv[i]) & (NXL - 1);
        const int ny = (y - eyv[i]) & (NYL - 1);
        f[i] = fin[((nx << 10) | ny) * 9 + i];
    }

    // ---- macroscopic + equilibrium moments (per lane, own cell) ----
    const float rho = ((f[0] + f[1]) + (f[2] + f[3])) +
                      ((f[4] + f[5]) + (f[6] + f[7])) + f[8];
    const float mx = f[1] - f[3] + f[5] - f[6] - f[7] + f[8];
    const float my = f[2] - f[4] + f[5] + f[6] - f[7] - f[8];
    // jx = mx + tau*GX*rho^2 (GX==0), jy = my + tau*GY*rho^2
    const float jx = mx;
    const float jy = my + (TAU * GYF) * rho * rho;
    const float jx2 = jx * jx, jy2 = jy * jy;
    const float j2 = jx2 + jy2;

    const float w0 = w[0], w1 = w[1], w2 = w[2], w3 = w[3];
    const float w4 = w[4], w5 = w[5], w6 = w[6];

    float me[9];
    me[0] = rho;
    me[1] = jx;
    me[2] = jy;
    me[3] = w0 * rho + w1 * j2;
    me[4] = w2 * rho - w3 * j2;
    me[5] = w4 * jx;
    me[6] = w4 * jy;
    me[7] = w5 * (jx2 - jy2);
    me[8] = w6 * jx * jy;

    // ---- A tiles (constants, lane-dependent) ----
    // A 16x4 layout: VGPR0 holds K=2*half, VGPR1 holds K=2*half+1, M=q16.
    const float* at = &g_AT.v[0][0];
    v2f a0, a1, a2;
    a0.x = at[(0 + 2 * half) * 16 + q16];
    a0.y = at[(1 + 2 * half) * 16 + q16];
    a1.x = at[(4 + 2 * half) * 16 + q16];
    a1.y = at[(5 + 2 * half) * 16 + q16];
    a2.x = at[(8 + 2 * half) * 16 + q16];
    a2.y = at[(9 + 2 * half) * 16 + q16];

    // ---- B tiles: B[k][cell], lower half supplies rows {0,1}, upper {2,3} ----
    v2f b0, b1, b2;
    b0.x = half ? me[2] : me[0];
    b0.y = half ? me[3] : me[1];
    b1.x = half ? me[6] : me[4];
    b1.y = half ? me[7] : me[5];
    b2.x = half ? 0.0f  : me[8];   // K rows 9..11 are zero padding
    b2.y = 0.0f;

    // ---- C += A*B over K = 0..11 : C[q][cell] = (Minv * meq / tau) ----
    v8f c = {};
    c = __builtin_amdgcn_wmma_f32_16x16x4_f32(false, a0, false, b0, (short)0, c,
                                              false, false);
    c = __builtin_amdgcn_wmma_f32_16x16x4_f32(false, a1, false, b1, (short)0, c,
                                              false, false);
    c = __builtin_amdgcn_wmma_f32_16x16x4_f32(false, a2, false, b2, (short)0, c,
                                              false, false);

    // ---- f' = (1 - 1/tau) * f_streamed + C ----
    const float decay = 1.0f - 1.0f / TAU;   // = -2/3
    float* o = fout + cell * 9;
    if (half == 0) {
        // lanes 0-15 hold C rows q = 0..7 of their own cell
#pragma unroll
        for (int q = 0; q < 8; ++q) o[q] = decay * f[q] + c[q];
    } else {
        // lanes 16-31 hold C rows q = 8..15; only q = 8 is valid
        o[8] = decay * f[8] + c[0];
    }
}

extern "C" void kernel_launch(void* const* d_in, const int* in_sizes, int n_in,
                              void* d_out, int out_size, void* d_ws, size_t ws_size,
                              hipStream_t stream) {
    const float* f0 = (const float*)d_in[0];   // [1024,1024,9] f32
    const float* w  = (const float*)d_in[1];   // [7] f32
    float* out = (float*)d_out;                // [1024,1024,9] f32
    float* ws  = (float*)d_ws;                 // >= 36 MB scratch

    // 2 threads per cell (16 cells per wave32) -> 2,097,152 threads
    const int threads = 256;
    const int blocks  = (NXL * NYL * 2) / threads;   // 8192

    // 4 global iterations with ping-pong (streaming needs global sync)
    lbm_step<<<blocks, threads, 0, stream>>>(f0,  w, ws);
    lbm_step<<<blocks, threads, 0, stream>>>(ws,  w, out);
    lbm_step<<<blocks, threads, 0, stream>>>(out, w, ws);
    lbm_step<<<blocks, threads, 0, stream>>>(ws,  w, out);
}